// FragmentGNN_32959579030068
// MI455X (gfx1250) — compile-verified
//
#include <hip/hip_runtime.h>

typedef float v2f __attribute__((ext_vector_type(2)));
typedef float v8f __attribute__((ext_vector_type(8)));

#define D 128

// ---------------- GEMM: H[N,128] = X[N,128] @ W[128,128] via fp32 WMMA ----
// Block = 256 threads = 8 waves; block handles one 16-row strip, each wave
// one 16-column tile (8*16 = 128 cols). K loop: 32 x V_WMMA_F32_16X16X4_F32.
__global__ __launch_bounds__(256) void gcn_gemm_wmma_f32(
    const float* __restrict__ X, const float* __restrict__ W,
    float* __restrict__ H)
{
    const int lane  = threadIdx.x & 31;
    const int wid   = threadIdx.x >> 5;        // 0..7 -> column tile
    const int m0    = blockIdx.x * 16;
    const int n0    = wid * 16;
    const int l15   = lane & 15;
    const int khalf = (lane >> 4) << 1;        // 0 (lanes 0-15) or 2 (16-31)

    // A: lane holds row m0+l15, cols k0+khalf .. k0+khalf+1  (8B aligned)
    const float* xrow = X + (size_t)(m0 + l15) * D + khalf;
    // B: lane holds col n0+l15, rows k0+khalf (b.x) and k0+khalf+1 (b.y)
    const float* wcol = W + (size_t)khalf * D + (n0 + l15);

    v8f acc = {};
#pragma unroll 8
    for (int k0 = 0; k0 < D; k0 += 4) {
        v2f a = *(const v2f*)(xrow + k0);
        v2f b;
        b.x = wcol[(size_t)k0 * D];
        b.y = wcol[(size_t)(k0 + 1) * D];
        acc = __builtin_amdgcn_wmma_f32_16x16x4_f32(
            false, a, false, b, (short)0, acc, false, false);
    }
    // C/D layout: VGPR j holds row m0 + j + 8*(lane>=16), col n0+l15
    float* hout = H + (size_t)(m0 + ((lane >> 4) << 3)) * D + n0 + l15;
#pragma unroll
    for (int j = 0; j < 8; ++j)
        hout[(size_t)j * D] = acc[j];
}

// ---------------- elementwise / graph helpers ----------------------------
__global__ void fill_f32(float* __restrict__ p, float v, int n) {
    int i = blockIdx.x * blockDim.x + threadIdx.x;
    if (i < n) p[i] = v;
}

__global__ void deg_accum(const int* __restrict__ dst, float* __restrict__ deg, int E) {
    int e = blockIdx.x * blockDim.x + threadIdx.x;
    if (e < E) atomicAdd(&deg[dst[e]], 1.0f);
}

__global__ void rsqrt_inplace(float* __restrict__ p, int n) {
    int i = blockIdx.x * blockDim.x + threadIdx.x;
    if (i < n) p[i] = rsqrtf(p[i]);
}

__global__ void norm_kernel(const int* __restrict__ src, const int* __restrict__ dst,
                            const float* __restrict__ dinv, float* __restrict__ nrm, int E) {
    int e = blockIdx.x * blockDim.x + threadIdx.x;
    if (e < E) nrm[e] = dinv[src[e]] * dinv[dst[e]];
}

// Y[i,:] = b + H[i,:] * dinv[i]^2   (self-loop term + bias), float4 per thread
__global__ void init_y(const float* __restrict__ H, const float* __restrict__ dinv,
                       const float* __restrict__ b, float* __restrict__ Y, int n32) {
    int t = blockIdx.x * blockDim.x + threadIdx.x;
    if (t >= n32) return;
    int i = t >> 5;
    int c = (t & 31) << 2;
    float s  = dinv[i];
    float ss = s * s;
    float4 h4 = *(const float4*)(H + (size_t)i * D + c);
    float4 b4 = *(const float4*)(b + c);
    float4 y;
    y.x = b4.x + h4.x * ss;
    y.y = b4.y + h4.y * ss;
    y.z = b4.z + h4.z * ss;
    y.w = b4.w + h4.w * ss;
    *(float4*)(Y + (size_t)i * D + c) = y;
}

// Y[dst[e],:] += H[src[e],:] * norm[e]   (32 threads x float4 per edge)
__global__ void scatter_edges(const int* __restrict__ src, const int* __restrict__ dst,
                              const float* __restrict__ nrm, const float* __restrict__ H,
                              float* __restrict__ Y, int e32) {
    int t = blockIdx.x * blockDim.x + threadIdx.x;
    if (t >= e32) return;
    int e = t >> 5;
    int c = (t & 31) << 2;
    float w = nrm[e];
    int s = src[e], d = dst[e];
    float4 h4 = *(const float4*)(H + (size_t)s * D + c);
    float* yp = Y + (size_t)d * D + c;
    atomicAdd(yp + 0, h4.x * w);
    atomicAdd(yp + 1, h4.y * w);
    atomicAdd(yp + 2, h4.z * w);
    atomicAdd(yp + 3, h4.w * w);
}

__global__ void relu_inplace(float* __restrict__ Y, int n32) {
    int t = blockIdx.x * blockDim.x + threadIdx.x;
    if (t >= n32) return;
    float4* p = (float4*)(Y + ((size_t)t << 2));
    float4 v = *p;
    v.x = fmaxf(v.x, 0.f); v.y = fmaxf(v.y, 0.f);
    v.z = fmaxf(v.z, 0.f); v.w = fmaxf(v.w, 0.f);
    *p = v;
}

__global__ void pool_cnt(const int* __restrict__ batch, float* __restrict__ cnt, int N) {
    int i = blockIdx.x * blockDim.x + threadIdx.x;
    if (i < N) atomicAdd(&cnt[batch[i]], 1.0f);
}

__global__ void pool_sum(const int* __restrict__ batch, const float* __restrict__ Y,
                         float* __restrict__ out, int n32) {
    int t = blockIdx.x * blockDim.x + threadIdx.x;
    if (t >= n32) return;
    int i = t >> 5;
    int c = (t & 31) << 2;
    int g = batch[i];
    float4 y4 = *(const float4*)(Y + (size_t)i * D + c);
    float* op = out + (size_t)g * D + c;
    atomicAdd(op + 0, y4.x);
    atomicAdd(op + 1, y4.y);
    atomicAdd(op + 2, y4.z);
    atomicAdd(op + 3, y4.w);
}

__global__ void pool_div(float* __restrict__ out, const float* __restrict__ cnt, int n) {
    int t = blockIdx.x * blockDim.x + threadIdx.x;
    if (t >= n) return;
    int g = t >> 7;   // D == 128
    out[t] /= fmaxf(cnt[g], 1.0f);
}

// ---------------- host driver --------------------------------------------
extern "C" void kernel_launch(void* const* d_in, const int* in_sizes, int n_in,
                              void* d_out, int out_size, void* d_ws, size_t ws_size,
                              hipStream_t stream) {
    const float* x     = (const float*)d_in[0];
    const int*   ei    = (const int*)d_in[1];
    const int*   batch = (const int*)d_in[2];
    const float* Ws[3] = {(const float*)d_in[3], (const float*)d_in[5], (const float*)d_in[7]};
    const float* bs[3] = {(const float*)d_in[4], (const float*)d_in[6], (const float*)d_in[8]};

    const int N = in_sizes[0] / D;     // 100000 (multiple of 16)
    const int E = in_sizes[1] / 2;     // 3200000
    const int G = out_size / D;        // 1024
    const int* src = ei;
    const int* dst = ei + E;

    float* ws = (float*)d_ws;
    size_t off = 0;
    float* dinv = ws + off; off += ((size_t)N + 255) & ~(size_t)255;
    float* nrm  = ws + off; off += ((size_t)E + 255) & ~(size_t)255;
    float* H    = ws + off; off += (size_t)N * D;
    float* Y    = ws + off; off += (size_t)N * D;
    float* cnt  = ws + off; off += (size_t)G;
    (void)ws_size; (void)n_in;

    const int B = 256;
    const int n32 = N * 32;            // N*D/4 float4 work-items
    const int e32 = E * 32;

    // deg -> dinv -> norm
    fill_f32<<<(N + B - 1) / B, B, 0, stream>>>(dinv, 1.0f, N);       // self-loop
    deg_accum<<<(E + B - 1) / B, B, 0, stream>>>(dst, dinv, E);
    rsqrt_inplace<<<(N + B - 1) / B, B, 0, stream>>>(dinv, N);
    norm_kernel<<<(E + B - 1) / B, B, 0, stream>>>(src, dst, dinv, nrm, E);

    const float* Xcur = x;
    for (int l = 0; l < 3; ++l) {
        gcn_gemm_wmma_f32<<<N / 16, 256, 0, stream>>>(Xcur, Ws[l], H);
        init_y<<<(n32 + B - 1) / B, B, 0, stream>>>(H, dinv, bs[l], Y, n32);
        scatter_edges<<<(e32 + B - 1) / B, B, 0, stream>>>(src, dst, nrm, H, Y, e32);
        relu_inplace<<<(n32 + B - 1) / B, B, 0, stream>>>(Y, n32);
        Xcur = Y;
    }

    // global mean pool
    float* out = (float*)d_out;
    fill_f32<<<(G * D + B - 1) / B, B, 0, stream>>>(out, 0.0f, G * D);
    fill_f32<<<(G + B - 1) / B, B, 0, stream>>>(cnt, 0.0f, G);
    pool_cnt<<<(N + B - 1) / B, B, 0, stream>>>(batch, cnt, N);
    pool_sum<<<(n32 + B - 1) / B, B, 0, stream>>>(batch, Y, out, n32);
    pool_div<<<(G * D + B - 1) / B, B, 0, stream>>>(out, cnt, G * D);
}